// ODEBlock_12987981103398
// MI455X (gfx1250) — compile-verified
//
#include <hip/hip_runtime.h>
#include <math.h>

// ---------------------------------------------------------------------------
// Neural-ODE (dopri5, 64 bounded iters) for f(y) = W3·relu(W2·relu(W1·y+b1)+b2)+b3
// Batch 16384, DIM=HID=256.  All 21 GEMMs per step are 16384x256x256 fp32.
// Compute-bound (~2.9 TFLOP fp32 total, all tensors L2-resident) ->
// V_WMMA_F32_16X16X4_F32 matrix path, fp32 end-to-end to keep the adaptive
// controller's accept decisions faithful to the reference.
// ---------------------------------------------------------------------------

typedef float v2f __attribute__((ext_vector_type(2)));
typedef float v8f __attribute__((ext_vector_type(8)));
typedef int   v4i __attribute__((ext_vector_type(4)));

#define BATCH_M 16384
#define NDIM    256
#define NELEM   (BATCH_M * NDIM)
#define NPART   512   // partial sums for the deterministic reduction

// CDNA5 async global->LDS staging (ASYNCcnt path), guarded so a missing
// builtin falls back to plain loads without breaking the build.
#if defined(__gfx1250__) && __has_builtin(__builtin_amdgcn_global_load_async_to_lds_b128)
#define USE_ASYNC 1
typedef __attribute__((address_space(1))) v4i* g_v4i_p;  // global src operand
typedef __attribute__((address_space(3))) v4i* l_v4i_p;  // LDS dst operand
#else
#define USE_ASYNC 0
#endif

#if __has_builtin(__builtin_amdgcn_s_wait_asynccnt)
#define WAIT_ASYNC0() __builtin_amdgcn_s_wait_asynccnt(0)
#else
#define WAIT_ASYNC0() asm volatile("s_wait_asynccnt 0x0" ::: "memory")
#endif

// ---------------------------------------------------------------------------
// Fused GEMM: C[M,256] = (A[M,256] @ W[256,256]) + bias, optional ReLU.
// Block = 256 threads (8 wave32), tile 128x64, K staged in LDS in 64-chunks.
// B is staged TRANSPOSED (Bt[col][k]) so both A and B fragments are single
// aligned ds_load_b64's.  Each wave owns a 32x32 slab = 4 v8f accumulators:
// 4 b64 LDS loads feed 4 WMMAs per K-step (1 DS-load per WMMA).
// ---------------------------------------------------------------------------
template <bool RELU>
__global__ __launch_bounds__(256) void gemm_wmma_f32(
    const float* __restrict__ A, const float* __restrict__ W,
    const float* __restrict__ bias, float* __restrict__ C) {
  constexpr int K = 256, N = 256, TM = 128, TN = 64, TK = 64;
  constexpr int LDA  = TK + 4;  // 68 floats/row: 4-bank stride -> conflict-free frags
  constexpr int LDBT = TK + 4;  // same for transposed B
  __shared__ float As[TM * LDA];
  __shared__ float Bt[TN * LDBT];

  const int tid  = threadIdx.x;
  const int wave = tid >> 5;
  const int lane = tid & 31;
  const int wm   = wave >> 1;        // 0..3 : 32-row slab inside tile
  const int wn   = wave & 1;         // 0..1 : 32-col slab inside tile
  const int row0 = blockIdx.y * TM;
  const int col0 = blockIdx.x * TN;

  // A loader: 16 rows x 16 float4 per pass, 8 passes
  const int lr = tid >> 4;           // 0..15
  const int lc = (tid & 15) << 2;    // 0,4,...,60
  // B loader: one thread per column, 16 k's per thread (coalesced b32 reads)
  const int bc = tid & 63;           // column 0..63
  const int bq = (tid >> 6) << 4;    // k-quarter base: 0,16,32,48

  // WMMA f32 16x16x4 fragment indexing (ISA 7.12.2):
  //   A: lanes 0-15 -> M=lane, K={kk,kk+1}; lanes 16-31 -> M=lane-16, K={kk+2,kk+3}
  //   B: lanes 0-15 -> N=lane, K rows {kk,kk+1}; lanes 16-31 -> K rows {kk+2,kk+3}
  const int m    = lane & 15;
  const int koff = (lane >> 4) << 1;

  v8f acc00 = {}, acc01 = {}, acc10 = {}, acc11 = {};

  for (int k0 = 0; k0 < K; k0 += TK) {
    // ---- stage A tile (128 x 64) ----
#pragma unroll
    for (int p = 0; p < 8; ++p) {
      const int r = (p << 4) + lr;
#if USE_ASYNC
      __builtin_amdgcn_global_load_async_to_lds_b128(
          (g_v4i_p)&A[(size_t)(row0 + r) * K + k0 + lc],
          (l_v4i_p)&As[r * LDA + lc], 0, 0);
#else
      *(float4*)&As[r * LDA + lc] =
          *(const float4*)&A[(size_t)(row0 + r) * K + k0 + lc];
#endif
    }
    // ---- stage B tile (64k x 64n) transposed into Bt[col][k] ----
#pragma unroll
    for (int i = 0; i < 16; i += 2) {
      float2 w;
      w.x = W[(size_t)(k0 + bq + i) * N + col0 + bc];
      w.y = W[(size_t)(k0 + bq + i + 1) * N + col0 + bc];
      *(float2*)&Bt[bc * LDBT + bq + i] = w;  // 8B aligned (even float offset)
    }
#if USE_ASYNC
    WAIT_ASYNC0();
#endif
    __syncthreads();

    const float* Arow0 = &As[(wm * 32 + m) * LDA + koff];
    const float* Arow1 = Arow0 + 16 * LDA;
    const float* Bcol0 = &Bt[(wn * 32 + m) * LDBT + koff];
    const float* Bcol1 = Bcol0 + 16 * LDBT;
#pragma unroll
    for (int kk = 0; kk < TK; kk += 4) {
      const v2f a0 = *(const v2f*)(Arow0 + kk);
      const v2f a1 = *(const v2f*)(Arow1 + kk);
      const v2f b0 = *(const v2f*)(Bcol0 + kk);
      const v2f b1 = *(const v2f*)(Bcol1 + kk);
      acc00 = __builtin_amdgcn_wmma_f32_16x16x4_f32(false, a0, false, b0,
                                                    (short)0, acc00, false, false);
      acc01 = __builtin_amdgcn_wmma_f32_16x16x4_f32(false, a0, false, b1,
                                                    (short)0, acc01, false, false);
      acc10 = __builtin_amdgcn_wmma_f32_16x16x4_f32(false, a1, false, b0,
                                                    (short)0, acc10, false, false);
      acc11 = __builtin_amdgcn_wmma_f32_16x16x4_f32(false, a1, false, b1,
                                                    (short)0, acc11, false, false);
    }
    __syncthreads();
  }

  // Epilogue: v8f layout -> element i: lanes0-15 row=i, lanes16-31 row=i+8
  const int half = lane >> 4;
#pragma unroll
  for (int i = 0; i < 8; ++i) {
    const int r0 = row0 + wm * 32 + i + half * 8;
    const int r1 = r0 + 16;
    const int c0 = col0 + wn * 32 + m;
    float v00 = acc00[i] + bias[c0];
    float v01 = acc01[i] + bias[c0 + 16];
    float v10 = acc10[i] + bias[c0];
    float v11 = acc11[i] + bias[c0 + 16];
    if (RELU) {
      v00 = fmaxf(v00, 0.f);
      v01 = fmaxf(v01, 0.f);
      v10 = fmaxf(v10, 0.f);
      v11 = fmaxf(v11, 0.f);
    }
    C[(size_t)r0 * NDIM + c0] = v00;
    C[(size_t)r0 * NDIM + c0 + 16] = v01;
    C[(size_t)r1 * NDIM + c0] = v10;
    C[(size_t)r1 * NDIM + c0 + 16] = v11;
  }
}

// ---------------------------------------------------------------------------
// Small helper kernels (state controller + elementwise stage algebra).
// ctl layout: [0]=t [1]=h [2]=h_eff [3]=done [4]=step_ok [5]=err_sum
// ---------------------------------------------------------------------------
__global__ void k_init_ctl(float* __restrict__ ctl) {
  if (threadIdx.x == 0 && blockIdx.x == 0) {
    ctl[0] = 0.0f;   // t
    ctl[1] = 0.05f;  // h = H_INIT
    ctl[2] = 0.0f;
    ctl[3] = 0.0f;
    ctl[4] = 0.0f;
    ctl[5] = 0.0f;
  }
}

__global__ void k_copy(const float* __restrict__ src, float* __restrict__ dst,
                       int n) {
  for (int i = blockIdx.x * blockDim.x + threadIdx.x; i < n;
       i += gridDim.x * blockDim.x)
    dst[i] = src[i];
}

__global__ void k_ctl_pre(float* __restrict__ ctl) {
  if (threadIdx.x == 0 && blockIdx.x == 0) {
    const float t = ctl[0], h = ctl[1];
    const bool done = t >= 1.0f - 1e-7f;
    ctl[2] = done ? h : fminf(h, 1.0f - t);
    ctl[3] = done ? 1.0f : 0.0f;
  }
}

// out = y + h_eff * (c1*p1 + ... + c_nk*p_nk)
__global__ void k_stage(const float* __restrict__ y, const float* __restrict__ p1,
                        const float* __restrict__ p2, const float* __restrict__ p3,
                        const float* __restrict__ p4, const float* __restrict__ p5,
                        float c1, float c2, float c3, float c4, float c5, int nk,
                        const float* __restrict__ ctl, float* __restrict__ out,
                        int n) {
  const float h = ctl[2];
  for (int i = blockIdx.x * blockDim.x + threadIdx.x; i < n;
       i += gridDim.x * blockDim.x) {
    float a = c1 * p1[i];
    if (nk > 1) a += c2 * p2[i];
    if (nk > 2) a += c3 * p3[i];
    if (nk > 3) a += c4 * p4[i];
    if (nk > 4) a += c5 * p5[i];
    out[i] = fmaf(h, a, y[i]);
  }
}

// per-element: err = h*(E1 k1 + E3 k3 + E4 k4 + E5 k5 + E6 k6 + E7 k7)
// r = err / (ATOL + RTOL*max(|y|,|y5|)); deterministic fixed-tree block sums.
__global__ __launch_bounds__(256) void k_reduce_err(
    const float* __restrict__ y, const float* __restrict__ y5v,
    const float* __restrict__ k1, const float* __restrict__ k3,
    const float* __restrict__ k4, const float* __restrict__ k5,
    const float* __restrict__ k6, const float* __restrict__ k7,
    const float* __restrict__ ctl, float* __restrict__ partial, int n) {
  const float E1 = (float)(71.0 / 57600.0);
  const float E3 = (float)(-71.0 / 16695.0);
  const float E4 = (float)(71.0 / 1920.0);
  const float E5 = (float)(-17253.0 / 339200.0);
  const float E6 = (float)(22.0 / 525.0);
  const float E7 = (float)(-1.0 / 40.0);
  const float h = ctl[2];
  float s = 0.0f;
  for (int i = blockIdx.x * blockDim.x + threadIdx.x; i < n;
       i += gridDim.x * blockDim.x) {
    float e = E1 * k1[i] + E3 * k3[i] + E4 * k4[i] + E5 * k5[i] + E6 * k6[i] +
              E7 * k7[i];
    e *= h;
    const float amax = fmaxf(fabsf(y[i]), fabsf(y5v[i]));
    const float tol = 1e-3f + 1e-3f * amax;
    const float r = e / tol;
    s += r * r;
  }
  __shared__ float red[256];
  red[threadIdx.x] = s;
  __syncthreads();
  for (int o = 128; o > 0; o >>= 1) {
    if (threadIdx.x < o) red[threadIdx.x] += red[threadIdx.x + o];
    __syncthreads();
  }
  if (threadIdx.x == 0) partial[blockIdx.x] = red[0];
}

__global__ __launch_bounds__(256) void k_reduce_fin(
    const float* __restrict__ partial, float* __restrict__ ctl) {
  __shared__ float red[256];
  red[threadIdx.x] = partial[threadIdx.x] + partial[threadIdx.x + 256];
  __syncthreads();
  for (int o = 128; o > 0; o >>= 1) {
    if (threadIdx.x < o) red[threadIdx.x] += red[threadIdx.x + o];
    __syncthreads();
  }
  if (threadIdx.x == 0) ctl[5] = red[0];
}

__global__ void k_ctl_post(float* __restrict__ ctl) {
  if (threadIdx.x == 0 && blockIdx.x == 0) {
    const float t = ctl[0], h = ctl[1], h_eff = ctl[2];
    const bool done = ctl[3] != 0.0f;
    const float err_norm = sqrtf(ctl[5] / (float)NELEM);
    const bool accept = err_norm <= 1.0f;
    float factor = 0.9f * powf(fmaxf(err_norm, 1e-10f), -0.2f);
    factor = fminf(fmaxf(factor, 0.2f), 10.0f);
    const bool step_ok = accept && !done;
    ctl[4] = step_ok ? 1.0f : 0.0f;
    if (step_ok) ctl[0] = t + h_eff;
    ctl[1] = done ? h : h_eff * factor;
  }
}

__global__ void k_accept(const float* __restrict__ ctl,
                         const float* __restrict__ ynew,
                         float* __restrict__ y, int n) {
  if (ctl[4] == 0.0f) return;  // uniform scalar branch across the grid
  for (int i = blockIdx.x * blockDim.x + threadIdx.x; i < n;
       i += gridDim.x * blockDim.x)
    y[i] = ynew[i];
}

// ---------------------------------------------------------------------------
extern "C" void kernel_launch(void* const* d_in, const int* in_sizes, int n_in,
                              void* d_out, int out_size, void* d_ws,
                              size_t ws_size, hipStream_t stream) {
  const float* x  = (const float*)d_in[0];
  const float* W1 = (const float*)d_in[1];
  const float* b1 = (const float*)d_in[2];
  const float* W2 = (const float*)d_in[3];
  const float* b2 = (const float*)d_in[4];
  const float* W3 = (const float*)d_in[5];
  const float* b3 = (const float*)d_in[6];
  float* y = (float*)d_out;  // evolving state lives in d_out

  float* ctl     = (float*)d_ws;
  float* partial = ctl + 64;
  float* base    = (float*)((char*)d_ws + 4096);
  float* zbuf = base + (size_t)0 * NELEM;
  float* h1b  = base + (size_t)1 * NELEM;
  float* h2b  = base + (size_t)2 * NELEM;
  float* k1   = base + (size_t)3 * NELEM;
  float* k2   = base + (size_t)4 * NELEM;  // reused as k7 after z6 is built
  float* k3   = base + (size_t)5 * NELEM;
  float* k4   = base + (size_t)6 * NELEM;
  float* k5   = base + (size_t)7 * NELEM;
  float* k6   = base + (size_t)8 * NELEM;
  float* y5b  = base + (size_t)9 * NELEM;
  float* k7   = k2;

  const dim3 gg(NDIM / 64, BATCH_M / 128);  // (4, 128) blocks
  const dim3 gb(256);
  const int EB = 2048, ET = 256;  // elementwise grid-stride config

  auto mlp = [&](const float* in, float* out) {
    gemm_wmma_f32<true><<<gg, gb, 0, stream>>>(in, W1, b1, h1b);
    gemm_wmma_f32<true><<<gg, gb, 0, stream>>>(h1b, W2, b2, h2b);
    gemm_wmma_f32<false><<<gg, gb, 0, stream>>>(h2b, W3, b3, out);
  };

  k_init_ctl<<<1, 32, 0, stream>>>(ctl);
  k_copy<<<EB, ET, 0, stream>>>(x, y, NELEM);

  for (int it = 0; it < 64; ++it) {
    k_ctl_pre<<<1, 32, 0, stream>>>(ctl);

    mlp(y, k1);
    k_stage<<<EB, ET, 0, stream>>>(y, k1, k1, k1, k1, k1,
                                   0.2f, 0.f, 0.f, 0.f, 0.f, 1, ctl, zbuf, NELEM);
    mlp(zbuf, k2);
    k_stage<<<EB, ET, 0, stream>>>(y, k1, k2, k1, k1, k1,
                                   (float)(3.0 / 40.0), (float)(9.0 / 40.0),
                                   0.f, 0.f, 0.f, 2, ctl, zbuf, NELEM);
    mlp(zbuf, k3);
    k_stage<<<EB, ET, 0, stream>>>(y, k1, k2, k3, k1, k1,
                                   (float)(44.0 / 45.0), (float)(-56.0 / 15.0),
                                   (float)(32.0 / 9.0), 0.f, 0.f, 3, ctl, zbuf,
                                   NELEM);
    mlp(zbuf, k4);
    k_stage<<<EB, ET, 0, stream>>>(y, k1, k2, k3, k4, k1,
                                   (float)(19372.0 / 6561.0),
                                   (float)(-25360.0 / 2187.0),
                                   (float)(64448.0 / 6561.0),
                                   (float)(-212.0 / 729.0), 0.f, 4, ctl, zbuf,
                                   NELEM);
    mlp(zbuf, k5);
    k_stage<<<EB, ET, 0, stream>>>(y, k1, k2, k3, k4, k5,
                                   (float)(9017.0 / 3168.0),
                                   (float)(-355.0 / 33.0),
                                   (float)(46732.0 / 5247.0),
                                   (float)(49.0 / 176.0),
                                   (float)(-5103.0 / 18656.0), 5, ctl, zbuf,
                                   NELEM);
    mlp(zbuf, k6);
    // y5 = y + h*(35/384 k1 + 500/1113 k3 + 125/192 k4 - 2187/6784 k5 + 11/84 k6)
    k_stage<<<EB, ET, 0, stream>>>(y, k1, k3, k4, k5, k6,
                                   (float)(35.0 / 384.0),
                                   (float)(500.0 / 1113.0),
                                   (float)(125.0 / 192.0),
                                   (float)(-2187.0 / 6784.0),
                                   (float)(11.0 / 84.0), 5, ctl, y5b, NELEM);
    mlp(y5b, k7);  // FSAL stage (overwrites k2, which is dead now)

    k_reduce_err<<<NPART, 256, 0, stream>>>(y, y5b, k1, k3, k4, k5, k6, k7, ctl,
                                            partial, NELEM);
    k_reduce_fin<<<1, 256, 0, stream>>>(partial, ctl);
    k_ctl_post<<<1, 32, 0, stream>>>(ctl);
    k_accept<<<EB, ET, 0, stream>>>(ctl, y5b, y, NELEM);
  }
}